// MambaMIM_53051436040362
// MI455X (gfx1250) — compile-verified
//
#include <hip/hip_runtime.h>

// ---------------------------------------------------------------------------
// MambaMIM mask-token reconstruction for MI455X (gfx1250, wave32, WMMA).
// B=4, C=256, F=8, N=512.
// Dominant work: Apow[k] = A^k, k=0..511 (17.1 GFLOP fp32) built with
// V_WMMA_F32_16X16X4_F32 via log-doubling (9 dependent batched GEMMs).
// The 128MB power table stays resident in the 192MB L2.
// GEMM: 4x M-tile register blocking per wave + async-to-LDS staging of the
// shared B panel (GLOBAL_LOAD_ASYNC_TO_LDS_B128 / ASYNCcnt on gfx1250).
// ---------------------------------------------------------------------------

#define B_DIM   4
#define C_DIM   256
#define N_TOK   512
#define MAT_EL  (C_DIM * C_DIM)   // 65536 floats per power matrix

typedef float v2f __attribute__((ext_vector_type(2)));
typedef float v8f __attribute__((ext_vector_type(8)));
typedef float v4f __attribute__((ext_vector_type(4)));
typedef int   v4i __attribute__((ext_vector_type(4)));

// addrspace-qualified pointer types for the async-to-LDS builtin:
//   param0: int4 addrspace(1)* (global), param1: int4 addrspace(3)* (LDS)
typedef __attribute__((address_space(1))) v4i* as1_v4i_p;
typedef __attribute__((address_space(3))) v4i* as3_v4i_p;

#if defined(__has_builtin)
#  if __has_builtin(__builtin_amdgcn_global_load_async_to_lds_b128)
#    define USE_ASYNC_LDS 1
#  endif
#endif
#ifndef USE_ASYNC_LDS
#  define USE_ASYNC_LDS 0
#endif

// ---------------------------------------------------------------------------
// Kernel 0: Apow[0] = I, Apow[1] = A
// ---------------------------------------------------------------------------
__global__ void init_powers(const float* __restrict__ A, float* __restrict__ apow) {
    int idx = blockIdx.x * blockDim.x + threadIdx.x;     // 0 .. 131071
    if (idx < MAT_EL) {
        int r = idx >> 8, c = idx & 255;
        apow[idx] = (r == c) ? 1.0f : 0.0f;              // A^0 = I
    } else {
        int j = idx - MAT_EL;
        apow[MAT_EL + j] = A[j];                          // A^1 = A
    }
}

// ---------------------------------------------------------------------------
// Kernel 1: power doubling GEMM.
//   Step `len`: P[len + 1 + m] = P[m + 1] @ P[len],  m = 0..nmat-1.
//   (inputs <= len, outputs > len: disjoint within a launch)
// Block = 256 threads = 8 waves; block owns a 256x32 output panel:
//   wave w: mgroup = w>>1 (64 rows = 4 M-tiles), tn_loc = w&1 (16 cols).
// B panel (256 rows x 32 cols = 32KB) staged in LDS via async-to-LDS;
// one B fragment then feeds 4 WMMAs (register blocking along M).
// grid.x = nmat * 8.
// ---------------------------------------------------------------------------
__global__ void power_gemm(float* __restrict__ apow, int len) {
    __shared__ float btile[C_DIM * 32];                   // 32 KB

    const int mat = blockIdx.x >> 3;                      // output matrix index m
    const int tn0 = (blockIdx.x & 7) * 2;                 // first of 2 tn strips
    const int c0  = tn0 * 16;                             // first column of panel

    const float* __restrict__ Amat = apow + (size_t)(mat + 1) * MAT_EL;   // P[m+1]
    const float* __restrict__ Bmat = apow + (size_t)len * MAT_EL;         // P[len]
    float*       __restrict__ Dmat = apow + (size_t)(len + 1 + mat) * MAT_EL;

    // ---- stage B[:, c0 .. c0+31] into LDS (2048 x 16B chunks, 8 per thread)
#pragma unroll
    for (int i = 0; i < 8; ++i) {
        int chunk = threadIdx.x + i * 256;                // 0..2047
        int r = chunk >> 3;                               // row 0..255
        int q = chunk & 7;                                // 16B piece within 128B
        const char* gsrc = (const char*)(Bmat + (size_t)r * C_DIM + c0) + q * 16;
        char*       ldst = (char*)(&btile[r * 32]) + q * 16;
#if USE_ASYNC_LDS
        __builtin_amdgcn_global_load_async_to_lds_b128(
            (as1_v4i_p)gsrc, (as3_v4i_p)ldst, 0, 0);
#else
        *(v4f*)ldst = *(const v4f*)gsrc;
#endif
    }
#if USE_ASYNC_LDS
#  if defined(__has_builtin) && __has_builtin(__builtin_amdgcn_s_wait_asynccnt)
    __builtin_amdgcn_s_wait_asynccnt(0);
#  else
    asm volatile("s_wait_asynccnt 0" ::: "memory");
#  endif
#endif
    __syncthreads();

    // ---- per-wave tile assignment
    const int w      = threadIdx.x >> 5;                  // 0..7 (wave32)
    const int lane   = threadIdx.x & 31;
    const int mg     = w >> 1;                            // 0..3 -> rows mg*64..+63
    const int tn_loc = w & 1;                             // which 16-col strip
    const int tn     = tn0 + tn_loc;
    const int half   = lane >> 4;                         // K/M split half
    const int l      = lane & 15;

    // A operand rows: lane l holds row M; VGPR0 = K(2*half), VGPR1 = K+1
    const float* arow[4];
#pragma unroll
    for (int j = 0; j < 4; ++j)
        arow[j] = Amat + (size_t)(mg * 64 + j * 16 + l) * C_DIM + 2 * half;

    const float* brow = &btile[tn_loc * 16 + l];          // + (K)*32 per step

    v8f acc[4] = {};

#pragma unroll 4
    for (int kk = 0; kk < C_DIM; kk += 4) {
        v2f b;
        b.x = brow[(kk + 2 * half) * 32];
        b.y = brow[(kk + 2 * half + 1) * 32];
#pragma unroll
        for (int j = 0; j < 4; ++j) {
            v2f a;
            a.x = arow[j][kk + 0];
            a.y = arow[j][kk + 1];
            // v_wmma_f32_16x16x4_f32 : D = A*B + C
            acc[j] = __builtin_amdgcn_wmma_f32_16x16x4_f32(
                /*neg_a=*/false, a, /*neg_b=*/false, b,
                /*c_mod=*/(short)0, acc[j], /*reuse_a=*/false, /*reuse_b=*/false);
        }
    }

    // C/D layout: VGPR r -> row M = r + 8*half, col N = lane&15
#pragma unroll
    for (int j = 0; j < 4; ++j) {
#pragma unroll
        for (int r = 0; r < 8; ++r) {
            int row = mg * 64 + j * 16 + r + half * 8;
            Dmat[(size_t)row * C_DIM + tn * 16 + l] = acc[j][r];
        }
    }
}

// ---------------------------------------------------------------------------
// Kernel 2: per-batch segment scan (serial, one lane per batch).
//   s[p]  = last kept index <= p          (forward cummax)
//   e[p]  = first kept index  > p  (or N) (reverse cummin, shifted)
//   alpha = (p-s)/max(gap-1,1) ; k = e-1-p ; overrides at p = N-1.
// ---------------------------------------------------------------------------
__global__ void seg_scan(const int* __restrict__ active,
                         int* __restrict__ s_arr, int* __restrict__ ec_arr,
                         float* __restrict__ alpha_arr, int* __restrict__ k_arr) {
    int b = threadIdx.x;
    if (b >= B_DIM) return;
    const int* fl = active + b * N_TOK;
    int* S  = s_arr  + b * N_TOK;
    int* EC = ec_arr + b * N_TOK;
    int* K  = k_arr  + b * N_TOK;
    float* AL = alpha_arr + b * N_TOK;

    // forward: s
    int cur = -1;
    for (int p = 0; p < N_TOK; ++p) {
        bool f = (p == 0 || p == N_TOK - 1) ? true : (fl[p] != 0);
        if (f) cur = p;
        S[p] = cur;
    }
    // backward: e = nxt[p+1], then alpha / k / ec
    int nxt = N_TOK;                       // nxt[N] = N
    for (int p = N_TOK - 1; p >= 0; --p) {
        bool f = (p == 0 || p == N_TOK - 1) ? true : (fl[p] != 0);
        int e = (p == N_TOK - 1) ? N_TOK : nxt;
        int s = S[p];
        int gap = e - s;
        int j = p - s;
        int gm1 = gap - 1; if (gm1 < 1) gm1 = 1;
        float alpha = (float)j / (float)gm1;
        int k = e - 1 - p;
        if (p == N_TOK - 1) { alpha = 0.0f; k = 1; }   // explicit final-token append
        EC[p] = (e < N_TOK - 1) ? e : (N_TOK - 1);
        AL[p] = alpha;
        K[p]  = k;
        if (f) nxt = p;                    // nxt[p] for next iteration
    }
}

// ---------------------------------------------------------------------------
// Kernel 3: AX[b][p][:] = Apow[k[p]] @ Xi,  Xi = (1-a)*X[s] + a*X[ec]
// grid = (N_TOK, B_DIM), block = 256 (one thread per output channel).
// X[p][d] = bcfff[b][d][p].
// ---------------------------------------------------------------------------
__global__ void matvec_ax(const float* __restrict__ bcfff,
                          const float* __restrict__ apow,
                          const int* __restrict__ s_arr, const int* __restrict__ ec_arr,
                          const float* __restrict__ alpha_arr, const int* __restrict__ k_arr,
                          float* __restrict__ AX) {
    __shared__ float xi[C_DIM];
    const int p = blockIdx.x;
    const int b = blockIdx.y;
    const int c = threadIdx.x;

    const int   sp = s_arr[b * N_TOK + p];
    const int   ec = ec_arr[b * N_TOK + p];
    const int   kk = k_arr[b * N_TOK + p];
    const float al = alpha_arr[b * N_TOK + p];

    const float xs = bcfff[((size_t)b * C_DIM + c) * N_TOK + sp];
    const float xe = bcfff[((size_t)b * C_DIM + c) * N_TOK + ec];
    xi[c] = (1.0f - al) * xs + al * xe;
    __syncthreads();

    const float* M = apow + (size_t)kk * MAT_EL + (size_t)c * C_DIM;
    float acc = 0.0f;
#pragma unroll 8
    for (int d = 0; d < C_DIM; ++d) acc = fmaf(M[d], xi[d], acc);

    AX[((size_t)b * N_TOK + p) * C_DIM + c] = acc;
}

// ---------------------------------------------------------------------------
// Kernel 4: per-batch cumsum + per-segment prefix subtraction + the torch-
// faithful (N,C)->(C,f,f,f) reinterpretation + where(active) select.
// grid = B_DIM blocks, block = 256 (one thread per channel).
// ---------------------------------------------------------------------------
__global__ void cumsum_out(const float* __restrict__ AX,
                           const int* __restrict__ s_arr,
                           const int* __restrict__ active,
                           const float* __restrict__ bcfff,
                           float* __restrict__ cs, float* __restrict__ tok,
                           float* __restrict__ out) {
    const int b = blockIdx.x;
    const int c = threadIdx.x;

    // full-axis inclusive cumsum per channel
    float acc = 0.0f;
    for (int p = 0; p < N_TOK; ++p) {
        acc += AX[((size_t)b * N_TOK + p) * C_DIM + c];
        cs[((size_t)b * N_TOK + p) * C_DIM + c] = acc;
    }
    __syncthreads();

    // per-segment cumsum = cs[p] - cs[s[p]-1]
    for (int p = 0; p < N_TOK; ++p) {
        int sp = s_arr[b * N_TOK + p];
        float pre = (sp > 0) ? cs[((size_t)b * N_TOK + (sp - 1)) * C_DIM + c] : 0.0f;
        tok[((size_t)b * N_TOK + p) * C_DIM + c] =
            cs[((size_t)b * N_TOK + p) * C_DIM + c] - pre;
    }
    __syncthreads();

    // mask_tok[b][c][n] = tok_flat[c*N + n] = tok[b][2c + n/256][n%256]
    for (int n = 0; n < N_TOK; ++n) {
        int p2 = 2 * c + (n >> 8);
        int d2 = n & 255;
        float mv   = tok[((size_t)b * N_TOK + p2) * C_DIM + d2];
        float orig = bcfff[((size_t)b * C_DIM + c) * N_TOK + n];
        out[((size_t)b * C_DIM + c) * N_TOK + n] = (active[b * N_TOK + n] != 0) ? orig : mv;
    }
}

// ---------------------------------------------------------------------------
// Host launcher. Inputs: d_in[0]=bcfff (f32), d_in[1]=cur_active (i32),
// d_in[2]=A (f32). Output: f32, B*C*N elements.
// ---------------------------------------------------------------------------
extern "C" void kernel_launch(void* const* d_in, const int* in_sizes, int n_in,
                              void* d_out, int out_size, void* d_ws, size_t ws_size,
                              hipStream_t stream) {
    const float* bcfff  = (const float*)d_in[0];
    const int*   active = (const int*)d_in[1];
    const float* A      = (const float*)d_in[2];
    float*       out    = (float*)d_out;

    // workspace layout (floats)
    float* apow = (float*)d_ws;                                // 512 * 65536
    float* AX   = apow + (size_t)N_TOK * MAT_EL;               // B*N*C
    float* cs   = AX   + (size_t)B_DIM * N_TOK * C_DIM;        // B*N*C
    float* tok  = cs   + (size_t)B_DIM * N_TOK * C_DIM;        // B*N*C
    int*   s_arr  = (int*)(tok + (size_t)B_DIM * N_TOK * C_DIM);
    int*   ec_arr = s_arr  + B_DIM * N_TOK;
    int*   k_arr  = ec_arr + B_DIM * N_TOK;
    float* alpha  = (float*)(k_arr + B_DIM * N_TOK);

    // Apow[0]=I, Apow[1]=A
    init_powers<<<dim3(2 * MAT_EL / 256), dim3(256), 0, stream>>>(A, apow);

    // log-doubling: 9 dependent batched GEMM launches build Apow[0..511]
    for (int len = 1; len <= 256; len <<= 1) {
        int nmat = (len < (N_TOK - 1) - len) ? len : (N_TOK - 1) - len;
        power_gemm<<<dim3(nmat * 8), dim3(256), 0, stream>>>(apow, len);
    }

    // segment metadata (tiny serial scans, one lane per batch)
    seg_scan<<<dim3(1), dim3(32), 0, stream>>>(active, s_arr, ec_arr, alpha, k_arr);

    // per-token matvec with Apow[k[p]]
    matvec_ax<<<dim3(N_TOK, B_DIM), dim3(256), 0, stream>>>(
        bcfff, apow, s_arr, ec_arr, alpha, k_arr, AX);

    // cumsum, segment prefix, reinterpret + where(active)
    cumsum_out<<<dim3(B_DIM), dim3(256), 0, stream>>>(
        AX, s_arr, active, bcfff, cs, tok, out);
}